// DWTModelAllBandsLearnable_39943195853367
// MI455X (gfx1250) — compile-verified
//
#include <hip/hip_runtime.h>
#include <cstdint>

typedef __attribute__((ext_vector_type(2))) float v2f;
typedef __attribute__((ext_vector_type(8))) float v8f;

#define INV_SQRT2 0.70710678118654752440f

// problem dims
#define BATCH 32
#define CH    3
#define H     512
#define W     512
#define TILE  64

// LDS tile geometry
#define XR  76   // x region rows/cols (64 + 6 halo each side)
#define XP  77   // padded stride
#define L1R 38   // level1 region (H/2 scale)
#define L1P 39
#define LLS 32   // LL1 interior region
#define LLP 33
#define L2P 17   // level2 16x16 padded
#define ZRR 18   // z region (H/4 scale), rows [J0-1, J0+17)
#define ZPP 19

// LDS float offsets (lifetime-based aliasing):
#define OFF_W    0                   // 2043 floats: dw729 | db9 | uw1296 | ub9
#define OFF_X    2048                // 3*76*77 = 17556
#define OFF_L1O  OFF_X               // alias: 9*32*33 = 9504
#define OFF_LLR  (OFF_X + 9504)      // alias: 3*32*33 = 3168
#define OFF_S1   (OFF_X + 17556)     // 9*38*39 = 13338
#define OFF_L2   (OFF_S1 + 13338)    // 12*16*17 = 3264
#define OFF_Z    (OFF_L2 + 3264)     // max(9*18*19=3078, 3*32*33=3168)
#define OFF_LL   OFF_Z
#define SMEM_FLOATS (OFF_Z + 3200)   // ~157.6 KB -> 2 workgroups / 320KB WGP

__global__ __launch_bounds__(256) void dwt_ae_fused_kernel(
    const float* __restrict__ x,  const float* __restrict__ dwn_w,
    const float* __restrict__ dwn_b, const float* __restrict__ up_w,
    const float* __restrict__ up_b, float* __restrict__ out)
{
  extern __shared__ float sm[];
  const int tid  = threadIdx.x;
  const int lane = tid & 31;
  const int wave = tid >> 5;

  const int b    = blockIdx.x >> 6;
  const int tile = blockIdx.x & 63;
  const int R0   = (tile >> 3) * TILE;   // full-res row base
  const int C0   = (tile & 7) * TILE;    // full-res col base
  const int J0   = R0 >> 2;              // H/4 row base of z tile
  const int Q0   = C0 >> 2;              // W/4 col base of z tile

  // ---- stage 0: weights -> LDS (tiny, plain loads) ----
  for (int i = tid; i < 729;  i += 256) sm[OFF_W + i]        = dwn_w[i];
  for (int i = tid; i < 9;    i += 256) sm[OFF_W + 729 + i]  = dwn_b[i];
  for (int i = tid; i < 1296; i += 256) sm[OFF_W + 738 + i]  = up_w[i];
  for (int i = tid; i < 9;    i += 256) sm[OFF_W + 2034 + i] = up_b[i];

  // ---- stage 1: x halo tile via ASYNC global->LDS DMA, clamped addresses ----
  // OOB halo garbage lands only where stage-2's `ok` mask forces zeros, so a
  // clamp (no EXEC divergence, no pre-zero pass) is sufficient.
  const float* xb = x + (size_t)b * CH * H * W;
  for (int i = tid; i < CH * XR * XR; i += 256) {
    int c = i / (XR * XR), r = (i / XR) % XR, q = i % XR;
    int gr = R0 - 6 + r;  gr = gr < 0 ? 0 : (gr > H - 1 ? H - 1 : gr);
    int gc = C0 - 6 + q;  gc = gc < 0 ? 0 : (gc > W - 1 ? W - 1 : gc);
    const float* gp = xb + ((size_t)c * H * W + (size_t)gr * W + gc);
    unsigned lds_addr = (unsigned)(uintptr_t)&sm[OFF_X + (c * XR + r) * XP + q];
    asm volatile("global_load_async_to_lds_b32 %0, %1, off"
                 :: "v"(lds_addr), "v"((unsigned long long)(uintptr_t)gp)
                 : "memory");
  }
  asm volatile("s_wait_asynccnt 0x0" ::: "memory");
  __syncthreads();

  // ---- stage 2: level-1 Haar: LH/HL/HH bands (38x38 w/ halo) + LL interior ----
  for (int i = tid; i < CH * L1R * L1R; i += 256) {
    int c = i / (L1R * L1R), li = (i / L1R) % L1R, lj = i % L1R;
    int gi = (R0 >> 1) - 3 + li, gj = (C0 >> 1) - 3 + lj;
    bool ok = (gi >= 0) && (gi < (H / 2)) && (gj >= 0) && (gj < (W / 2));
    const float* p = &sm[OFF_X + (c * XR + 2 * li) * XP + 2 * lj];
    float a = p[0], bb = p[1], cc = p[XP], dd = p[XP + 1];
    float aw0 = (a + bb) * INV_SQRT2, dw0 = (a - bb) * INV_SQRT2;
    float aw1 = (cc + dd) * INV_SQRT2, dw1 = (cc - dd) * INV_SQRT2;
    float LL = ok ? (aw0 + aw1) * INV_SQRT2 : 0.f;
    float LH = ok ? (aw0 - aw1) * INV_SQRT2 : 0.f;
    float HL = ok ? (dw0 + dw1) * INV_SQRT2 : 0.f;
    float HH = ok ? (dw0 - dw1) * INV_SQRT2 : 0.f;
    sm[OFF_S1 + ((c * 3 + 0) * L1R + li) * L1P + lj] = LH;
    sm[OFF_S1 + ((c * 3 + 1) * L1R + li) * L1P + lj] = HL;
    sm[OFF_S1 + ((c * 3 + 2) * L1R + li) * L1P + lj] = HH;
    if (li >= 3 && li < 35 && lj >= 3 && lj < 35)
      sm[OFF_LL + (c * LLS + (li - 3)) * LLP + (lj - 3)] = LL;
  }
  __syncthreads();

  // ---- stage 3: level-2 Haar from LL1 (32x32 -> 4 bands of 16x16) ----
  for (int i = tid; i < CH * 16 * 16; i += 256) {
    int c = i >> 8, r2 = (i >> 4) & 15, c2 = i & 15;
    const float* p = &sm[OFF_LL + (c * LLS + 2 * r2) * LLP + 2 * c2];
    float a = p[0], bb = p[1], cc = p[LLP], dd = p[LLP + 1];
    float aw0 = (a + bb) * INV_SQRT2, dw0 = (a - bb) * INV_SQRT2;
    float aw1 = (cc + dd) * INV_SQRT2, dw1 = (cc - dd) * INV_SQRT2;
    sm[OFF_L2 + ((c * 4 + 0) * 16 + r2) * L2P + c2] = (aw0 + aw1) * INV_SQRT2;
    sm[OFF_L2 + ((c * 4 + 1) * 16 + r2) * L2P + c2] = (aw0 - aw1) * INV_SQRT2;
    sm[OFF_L2 + ((c * 4 + 2) * 16 + r2) * L2P + c2] = (dw0 + dw1) * INV_SQRT2;
    sm[OFF_L2 + ((c * 4 + 3) * 16 + r2) * L2P + c2] = (dw0 - dw1) * INV_SQRT2;
  }
  __syncthreads();

  // ---- stage 4: conv 9->9 3x3 s2 p1 as implicit GEMM on WMMA f32 16x16x4 ----
  // M = 324 z positions (21 tiles), N = 9 (pad16), K = 81 (21 steps of 4).
  // All offsets / B fragments are m-tile invariant -> hoisted into registers.
  // Zero-padding (k>=81) flows through B only; out-of-range M rows are never
  // stored, so the MAC loop carries no masks and no divergence.
  {
    const int mloc  = lane & 15;   // A: M row | B,D: N col
    const int khalf = lane >> 4;   // K pair select
    int   aoff[21][2];
    float bfv [21][2];
#pragma unroll
    for (int kk = 0; kk < 21; ++kk) {
#pragma unroll
      for (int t = 0; t < 2; ++t) {
        int k = kk * 4 + 2 * khalf + t;
        if (k < 81) {
          int ic = k / 9, tap = k % 9, kh = tap / 3, kw = tap % 3;
          aoff[kk][t] = (ic * L1R + kh) * L1P + kw;
          bfv[kk][t]  = (mloc < 9) ? sm[OFF_W + mloc * 81 + k] : 0.f;
        } else {
          aoff[kk][t] = 0;
          bfv[kk][t]  = 0.f;
        }
      }
    }
    for (int mt = wave; mt < 21; mt += 8) {
      int m  = mt * 16 + mloc;
      int mc = m < 324 ? m : 323;            // clamp: keeps reads in-range
      int zr = mc / ZRR, zc = mc % ZRR;
      const float* baseA = &sm[OFF_S1 + (2 * zr) * L1P + 2 * zc];
      v8f acc = {0.f, 0.f, 0.f, 0.f, 0.f, 0.f, 0.f, 0.f};
#pragma unroll
      for (int kk = 0; kk < 21; ++kk) {
        v2f af, bf;
        af[0] = baseA[aoff[kk][0]];
        af[1] = baseA[aoff[kk][1]];
        bf[0] = bfv[kk][0];
        bf[1] = bfv[kk][1];
        acc = __builtin_amdgcn_wmma_f32_16x16x4_f32(
            false, af, false, bf, (short)0, acc, false, false);
      }
#pragma unroll
      for (int dv = 0; dv < 8; ++dv) {
        int mm = mt * 16 + dv + 8 * khalf;   // D row = spatial position
        if (mloc < 9 && mm < 324) {
          int zr2 = mm / ZRR, zc2 = mm % ZRR;
          int jg = J0 - 1 + zr2, cg = Q0 - 1 + zc2;
          float val = 0.f;
          if (jg >= 0 && jg < (H / 4) && cg >= 0 && cg < (W / 4))
            val = acc[dv] + sm[OFF_W + 729 + mloc];
          sm[OFF_Z + (mloc * ZRR + zr2) * ZPP + zc2] = val;  // masked OOB -> 0
        }
      }
    }
  }
  __syncthreads();

  // ---- stage 5: convT 9->9 4x4 s2 p1, 4 parity classes, WMMA f32 16x16x4 ----
  // per class: fixed 2x2 taps -> K = 9ic*4 (9 steps); B frags preloaded/class.
  {
    const int mloc  = lane & 15;
    const int khalf = lane >> 4;   // == th (row tap select)
#pragma unroll
    for (int cls = 0; cls < 4; ++cls) {
      int pr = cls >> 1, pc = cls & 1;
      float bfv2[9][2];
#pragma unroll
      for (int ic = 0; ic < 9; ++ic) {
#pragma unroll
        for (int t = 0; t < 2; ++t) {       // t == tw (col tap select)
          int kh = 3 - pr - 2 * khalf, kw = 3 - pc - 2 * t;
          bfv2[ic][t] = (mloc < 9)
              ? sm[OFF_W + 738 + ((ic * 9 + mloc) * 4 + kh) * 4 + kw]
              : 0.f;
        }
      }
      for (int mt = wave; mt < 16; mt += 8) {
        const float* baseZ =
            &sm[OFF_Z + (mt + khalf + pr) * ZPP + mloc + pc];
        v8f acc = {0.f, 0.f, 0.f, 0.f, 0.f, 0.f, 0.f, 0.f};
#pragma unroll
        for (int ic = 0; ic < 9; ++ic) {
          v2f af, bf;
          af[0] = baseZ[ic * ZRR * ZPP];
          af[1] = baseZ[ic * ZRR * ZPP + 1];
          bf[0] = bfv2[ic][0];
          bf[1] = bfv2[ic][1];
          acc = __builtin_amdgcn_wmma_f32_16x16x4_f32(
              false, af, false, bf, (short)0, acc, false, false);
        }
#pragma unroll
        for (int dv = 0; dv < 8; ++dv) {
          if (mloc < 9) {                    // N = out channel
            int lc2o = dv + 8 * khalf;       // D row = col position in tile
            sm[OFF_L1O + (mloc * 32 + (2 * mt + pr)) * 33 + (2 * lc2o + pc)] =
                acc[dv] + sm[OFF_W + 2034 + mloc];
          }
        }
      }
    }
  }
  __syncthreads();

  // ---- stage 6: inverse Haar level 2: l2 bands -> LL1r (32x32) ----
  for (int i = tid; i < CH * 16 * 16; i += 256) {
    int c = i >> 8, r2 = (i >> 4) & 15, c2 = i & 15;
    float LL = sm[OFF_L2 + ((c * 4 + 0) * 16 + r2) * L2P + c2];
    float LH = sm[OFF_L2 + ((c * 4 + 1) * 16 + r2) * L2P + c2];
    float HL = sm[OFF_L2 + ((c * 4 + 2) * 16 + r2) * L2P + c2];
    float HH = sm[OFF_L2 + ((c * 4 + 3) * 16 + r2) * L2P + c2];
    float awe = (LL + LH) * INV_SQRT2, awo = (LL - LH) * INV_SQRT2;
    float dwe = (HL + HH) * INV_SQRT2, dwo = (HL - HH) * INV_SQRT2;
    float* q = &sm[OFF_LLR + (c * 32 + 2 * r2) * 33 + 2 * c2];
    q[0]  = (awe + dwe) * INV_SQRT2;
    q[1]  = (awe - dwe) * INV_SQRT2;
    q[33] = (awo + dwo) * INV_SQRT2;
    q[34] = (awo - dwo) * INV_SQRT2;
  }
  __syncthreads();

  // ---- stage 7: inverse Haar level 1 -> global output (float2 stores) ----
  float* ob = out + (size_t)b * CH * H * W;
  for (int i = tid; i < CH * 32 * 32; i += 256) {
    int c = i >> 10, r = (i >> 5) & 31, q = i & 31;
    float LL = sm[OFF_LLR + (c * 32 + r) * 33 + q];
    float LH = sm[OFF_L1O + ((c * 3 + 0) * 32 + r) * 33 + q];
    float HL = sm[OFF_L1O + ((c * 3 + 1) * 32 + r) * 33 + q];
    float HH = sm[OFF_L1O + ((c * 3 + 2) * 32 + r) * 33 + q];
    float awe = (LL + LH) * INV_SQRT2, awo = (LL - LH) * INV_SQRT2;
    float dwe = (HL + HH) * INV_SQRT2, dwo = (HL - HH) * INV_SQRT2;
    int gr = R0 + 2 * r, gc = C0 + 2 * q;
    size_t base = (size_t)c * H * W + (size_t)gr * W + gc;
    float2 top, bot;
    top.x = (awe + dwe) * INV_SQRT2;  top.y = (awe - dwe) * INV_SQRT2;
    bot.x = (awo + dwo) * INV_SQRT2;  bot.y = (awo - dwo) * INV_SQRT2;
    *(float2*)(ob + base)     = top;   // gc even -> 8B aligned
    *(float2*)(ob + base + W) = bot;
  }
}

extern "C" void kernel_launch(void* const* d_in, const int* in_sizes, int n_in,
                              void* d_out, int out_size, void* d_ws, size_t ws_size,
                              hipStream_t stream) {
  const float* x  = (const float*)d_in[0];
  const float* dw = (const float*)d_in[1];
  const float* db = (const float*)d_in[2];
  const float* uw = (const float*)d_in[3];
  const float* ub = (const float*)d_in[4];
  float* out = (float*)d_out;

  dim3 grid(BATCH * 64);          // 8x8 tiles of 64x64 per image, 32 images
  dim3 block(256);                // 8 wave32 waves
  size_t shmem = (size_t)SMEM_FLOATS * sizeof(float);
  dwt_ae_fused_kernel<<<grid, block, shmem, stream>>>(x, dw, db, uw, ub, out);
}